// MixHopNetwork_25606595018871
// MI455X (gfx1250) — compile-verified
//
#include <hip/hip_runtime.h>
#include <math.h>

#define Nn    100000
#define Ee    3200000
#define NNZn  1600000
#define FIN   512
#define Bq    4096
#define H1n   64
#define H2n   32

typedef __attribute__((ext_vector_type(2))) float v2f;
typedef __attribute__((ext_vector_type(8))) float v8f;

__device__ __forceinline__ float elu1(float x) {
    return x > 0.f ? x : (expf(x) - 1.f);
}

// ---------------- utility ----------------
__global__ void k_zero(float* __restrict__ p, int n) {
    int t = blockIdx.x * blockDim.x + threadIdx.x;
    if (t < n) p[t] = 0.f;
}

// out[r*os+oo+k] += vals[e] * x[c*xs+k]   (wave-per-edge, lane-per-column, 32 cols)
__global__ void k_spmm32(const int* __restrict__ rows, const int* __restrict__ cols,
                         const float* __restrict__ vals, const float* __restrict__ x,
                         int xs, float* __restrict__ out, int os, int oo, int nnz) {
    int t = blockIdx.x * blockDim.x + threadIdx.x;
    int e = t >> 5;
    if (e >= nnz) return;
    int k = t & 31;
    int r = rows[e];
    int c = cols[e];
    float v = vals[e];
    atomicAdd(&out[r * os + oo + k], v * x[c * xs + k]);
}

__global__ void k_bias_relu(const float* __restrict__ in, const float* __restrict__ b,
                            float* __restrict__ out, int os, int oo, int nrow) {
    int t = blockIdx.x * blockDim.x + threadIdx.x;
    if (t >= nrow * 32) return;
    int r = t >> 5, k = t & 31;
    float v = in[t] + b[k];
    out[r * os + oo + k] = v > 0.f ? v : 0.f;
}

__global__ void k_bias_add(float* __restrict__ out, const float* __restrict__ b,
                           int os, int oo, int nrow) {
    int t = blockIdx.x * blockDim.x + threadIdx.x;
    if (t >= nrow * 32) return;
    int r = t >> 5, k = t & 31;
    out[r * os + oo + k] += b[k];
}

// pack w_bot[4,128,32] -> Wt[(i*32+j)*128 + k]  (transposed: Wt[n][k])
__global__ void k_pack_wbot(const float* __restrict__ wb, float* __restrict__ Wt) {
    int t = blockIdx.x * blockDim.x + threadIdx.x;
    if (t >= 4 * 128 * 32) return;
    int i = t >> 12;
    int rem = t & 4095;
    int k = rem >> 5;
    int j = rem & 31;
    Wt[(i * 32 + j) * 128 + k] = wb[i * 4096 + k * 32 + j];
}

// ---------------- dense GEMM: C[M,*] = A[M,128] x Bt^T, Bt is [32,128] row-major (Bt[n][k])
__global__ void k_gemm_nk128(const float* __restrict__ A, const float* __restrict__ Bt,
                             float* __restrict__ C, int ldc, int coff, int Mrows) {
    int wave = blockIdx.x * (blockDim.x >> 5) + (threadIdx.x >> 5);
    int rowtile = wave >> 1;
    int coltile = wave & 1;
    if (rowtile * 16 >= Mrows) return;           // wave-uniform guard: EXEC all-ones in WMMA
    int lane = threadIdx.x & 31;
    int l15 = lane & 15;
    int hi  = lane >> 4;
    const float* Ap = A  + (rowtile * 16 + l15) * 128 + hi * 2;
    const float* Bp = Bt + (coltile * 16 + l15) * 128 + hi * 2;
    v8f acc = {0.f, 0.f, 0.f, 0.f, 0.f, 0.f, 0.f, 0.f};
#pragma unroll
    for (int k0 = 0; k0 < 128; k0 += 4) {
        v2f a = *(const v2f*)(Ap + k0);
        v2f b = *(const v2f*)(Bp + k0);
        acc = __builtin_amdgcn_wmma_f32_16x16x4_f32(false, a, false, b, (short)0, acc,
                                                    false, false);
    }
    float* Cp = C + coff + coltile * 16 + l15;
    int rbase = rowtile * 16 + 8 * hi;
#pragma unroll
    for (int r = 0; r < 8; ++r) Cp[(rbase + r) * ldc] = acc[r];
}

// ---------------- head ----------------
__global__ void k_gather(const float* __restrict__ latent, const int* __restrict__ idx,
                         float* __restrict__ p1, float* __restrict__ p2) {
    int t = blockIdx.x * blockDim.x + threadIdx.x;
    if (t >= Bq * 128) return;
    int b = t >> 7, k = t & 127;
    p1[t] = latent[idx[b] * 128 + k];
    p2[t] = latent[idx[Bq + b] * 128 + k];
}

// feat[b,h] = elu( sum_ij p1[b,i]*blw[h,i,j]*p2[b,j] + blb[h] )
// wave per (h, 16-row b-tile); 8 column tiles of 16; K=128 per tile via WMMA.
__global__ void k_bilinear(const float* __restrict__ p1, const float* __restrict__ p2,
                           const float* __restrict__ blw, const float* __restrict__ blb,
                           float* __restrict__ feat) {
    int h  = blockIdx.y * 8 + (threadIdx.x >> 5);
    int b0 = blockIdx.x * 16;
    int lane = threadIdx.x & 31;
    int l15 = lane & 15;
    int hi  = lane >> 4;
    const float* Ap = p1 + (b0 + l15) * 128 + hi * 2;
    const float* W  = blw + h * (128 * 128);
    float racc[8] = {0.f, 0.f, 0.f, 0.f, 0.f, 0.f, 0.f, 0.f};
    for (int nt = 0; nt < 8; ++nt) {
        const float* Bp = W + (hi * 2) * 128 + nt * 16 + l15;   // W[(k+2hi)*128 + n]
        v8f acc = {0.f, 0.f, 0.f, 0.f, 0.f, 0.f, 0.f, 0.f};
#pragma unroll
        for (int k0 = 0; k0 < 128; k0 += 4) {
            v2f a = *(const v2f*)(Ap + k0);
            v2f b;
            b.x = Bp[k0 * 128];
            b.y = Bp[(k0 + 1) * 128];
            acc = __builtin_amdgcn_wmma_f32_16x16x4_f32(false, a, false, b, (short)0, acc,
                                                        false, false);
        }
        int nglob = nt * 16 + l15;
#pragma unroll
        for (int r = 0; r < 8; ++r)
            racc[r] += acc[r] * p2[(b0 + 8 * hi + r) * 128 + nglob];
    }
    // reduce over the 16 N-lanes inside each half-wave
#pragma unroll
    for (int r = 0; r < 8; ++r) {
        float v = racc[r];
        v += __shfl_xor(v, 1, 32);
        v += __shfl_xor(v, 2, 32);
        v += __shfl_xor(v, 4, 32);
        v += __shfl_xor(v, 8, 32);
        racc[r] = v;
    }
    if (l15 == 0) {
        float bb = blb[h];
#pragma unroll
        for (int r = 0; r < 8; ++r)
            feat[(b0 + 8 * hi + r) * H1n + h] = elu1(racc[r] + bb);
    }
}

// z1 = elu(feat[4096,64] @ d1w[64,32] + d1b)
__global__ void k_dense1(const float* __restrict__ feat, const float* __restrict__ w,
                         const float* __restrict__ bias, float* __restrict__ z) {
    int wave = blockIdx.x * 8 + (threadIdx.x >> 5);
    int rowtile = wave >> 1;
    int coltile = wave & 1;
    if (rowtile >= Bq / 16) return;
    int lane = threadIdx.x & 31;
    int l15 = lane & 15;
    int hi  = lane >> 4;
    const float* Ap = feat + (rowtile * 16 + l15) * 64 + hi * 2;
    int nglob = coltile * 16 + l15;
    v8f acc = {0.f, 0.f, 0.f, 0.f, 0.f, 0.f, 0.f, 0.f};
#pragma unroll
    for (int k0 = 0; k0 < 64; k0 += 4) {
        v2f a = *(const v2f*)(Ap + k0);
        v2f b;
        b.x = w[(k0 + hi * 2) * 32 + nglob];
        b.y = w[(k0 + hi * 2 + 1) * 32 + nglob];
        acc = __builtin_amdgcn_wmma_f32_16x16x4_f32(false, a, false, b, (short)0, acc,
                                                    false, false);
    }
    float bb = bias[nglob];
#pragma unroll
    for (int r = 0; r < 8; ++r) {
        int row = rowtile * 16 + 8 * hi + r;
        z[row * 32 + nglob] = elu1(acc[r] + bb);
    }
}

__global__ void k_dense2(const float* __restrict__ z, const float* __restrict__ w,
                         const float* __restrict__ b, float* __restrict__ pred) {
    int t = blockIdx.x * blockDim.x + threadIdx.x;
    if (t >= Bq) return;
    float s = 0.f;
#pragma unroll
    for (int k = 0; k < 32; ++k) s += z[t * 32 + k] * w[k];
    pred[t] = s + b[0];
}

// ---------------- host orchestration ----------------
extern "C" void kernel_launch(void* const* d_in, const int* in_sizes, int n_in,
                              void* d_out, int out_size, void* d_ws, size_t ws_size,
                              hipStream_t stream) {
    (void)in_sizes; (void)n_in; (void)ws_size;
    const int*   adj_rows = (const int*)  d_in[0];
    const int*   adj_cols = (const int*)  d_in[1];
    const float* adj_vals = (const float*)d_in[2];
    const int*   f_rows   = (const int*)  d_in[3];
    const int*   f_cols   = (const int*)  d_in[4];
    const float* f_vals   = (const float*)d_in[5];
    const int*   idx      = (const int*)  d_in[6];
    const float* w_up     = (const float*)d_in[7];
    const float* b_up     = (const float*)d_in[8];
    const float* w_bot    = (const float*)d_in[9];
    const float* b_bot    = (const float*)d_in[10];
    const float* blw      = (const float*)d_in[11];
    const float* blb      = (const float*)d_in[12];
    const float* d1w      = (const float*)d_in[13];
    const float* d1b      = (const float*)d_in[14];
    const float* d2w      = (const float*)d_in[15];
    const float* d2b      = (const float*)d_in[16];

    float* out    = (float*)d_out;
    float* latent = out + Bq;                 // output tuple: [pred(4096) | latent(N*128)]

    float* af1  = (float*)d_ws;               // N*128
    float* U    = af1  + (size_t)Nn * 128;    // N*32 ping
    float* V    = U    + (size_t)Nn * 32;     // N*32 pong
    float* Wt   = V    + (size_t)Nn * 32;     // 128*128 packed w_bot^T
    float* feat = Wt   + 128 * 128;           // B*H1
    float* z1   = feat + (size_t)Bq * H1n;    // B*H2
    float* p1   = z1   + (size_t)Bq * H2n;    // B*128
    float* p2   = p1   + (size_t)Bq * 128;    // B*128

    auto zero = [&](float* p, int n) {
        k_zero<<<(n + 255) / 256, 256, 0, stream>>>(p, n);
    };

    zero(out, out_size);                      // predictions + latent (atomic targets)
    zero(af1, Nn * 128);
    k_pack_wbot<<<(4 * 128 * 32 + 255) / 256, 256, 0, stream>>>(w_bot, Wt);

    const int spmmF_blocks = (NNZn * 32) / 256;   // exact
    const int spmmA_blocks = (Ee * 32) / 256;     // exact
    const int nb32 = (Nn * 32 + 255) / 256;

    // ---- upper SparseNGCN layers: af1[:, 32i:32i+32] = hops^i( relu(spmm(feat, w_up[i]) + b_up[i]) )
    for (int i = 0; i < 4; ++i) {
        zero(U, Nn * 32);
        k_spmm32<<<spmmF_blocks, 256, 0, stream>>>(f_rows, f_cols, f_vals,
                                                   w_up + i * FIN * 32, 32, U, 32, 0, NNZn);
        if (i == 0) {
            k_bias_relu<<<nb32, 256, 0, stream>>>(U, b_up, af1, 128, 0, Nn);
        } else {
            k_bias_relu<<<nb32, 256, 0, stream>>>(U, b_up + i * 32, U, 32, 0, Nn);
            float* cur = U; float* oth = V;
            for (int h = 0; h < i; ++h) {
                if (h == i - 1) {
                    k_spmm32<<<spmmA_blocks, 256, 0, stream>>>(adj_rows, adj_cols, adj_vals,
                                                               cur, 32, af1, 128, i * 32, Ee);
                } else {
                    zero(oth, Nn * 32);
                    k_spmm32<<<spmmA_blocks, 256, 0, stream>>>(adj_rows, adj_cols, adj_vals,
                                                               cur, 32, oth, 32, 0, Ee);
                    float* t = cur; cur = oth; oth = t;
                }
            }
        }
    }

    // ---- bottom DenseNGCN layers: latent[:, 32i:32i+32] = hops^i(af1 @ w_bot[i]) + b_bot[i]
    const int gemmBlocks = ((Nn / 16) * 2 + 7) / 8;
    for (int i = 0; i < 4; ++i) {
        if (i == 0) {
            k_gemm_nk128<<<gemmBlocks, 256, 0, stream>>>(af1, Wt, latent, 128, 0, Nn);
        } else {
            k_gemm_nk128<<<gemmBlocks, 256, 0, stream>>>(af1, Wt + i * 32 * 128, U, 32, 0, Nn);
            float* cur = U; float* oth = V;
            for (int h = 0; h < i; ++h) {
                if (h == i - 1) {
                    k_spmm32<<<spmmA_blocks, 256, 0, stream>>>(adj_rows, adj_cols, adj_vals,
                                                               cur, 32, latent, 128, i * 32, Ee);
                } else {
                    zero(oth, Nn * 32);
                    k_spmm32<<<spmmA_blocks, 256, 0, stream>>>(adj_rows, adj_cols, adj_vals,
                                                               cur, 32, oth, 32, 0, Ee);
                    float* t = cur; cur = oth; oth = t;
                }
            }
        }
        k_bias_add<<<nb32, 256, 0, stream>>>(latent, b_bot + i * 32, 128, i * 32, Nn);
    }

    // ---- pair head ----
    k_gather<<<(Bq * 128) / 256, 256, 0, stream>>>(latent, idx, p1, p2);
    k_bilinear<<<dim3(Bq / 16, 8), 256, 0, stream>>>(p1, p2, blw, blb, feat);
    k_dense1<<<((Bq / 16) * 2 + 7) / 8, 256, 0, stream>>>(feat, d1w, d1b, z1);
    k_dense2<<<(Bq + 255) / 256, 256, 0, stream>>>(z1, d2w, d2b, out);
}